// AttentiveMLP2_6236292513984
// MI455X (gfx1250) — compile-verified
//
#include <hip/hip_runtime.h>
#include <hip/hip_bf16.h>

typedef __attribute__((ext_vector_type(2))) float v2f;
typedef __attribute__((ext_vector_type(8))) float v8f;

#define N_NODES 100000
#define N_EDGES 1600000
#define DFEAT   128

__device__ __forceinline__ v8f wmma_f32_k4(v2f a, v2f b, v8f c) {
    // V_WMMA_F32_16X16X4_F32 : D = A(16x4) x B(4x16) + C(16x16), wave32
    return __builtin_amdgcn_wmma_f32_16x16x4_f32(
        /*neg_a=*/false, a, /*neg_b=*/false, b,
        /*c_mod=*/(short)0, c, /*reuse_a=*/false, /*reuse_b=*/false);
}

// ---- order-preserving float <-> uint mapping for atomic segment-max --------
__device__ __forceinline__ unsigned f32_to_ord(float f) {
    unsigned u = __float_as_uint(f);
    return (u & 0x80000000u) ? ~u : (u | 0x80000000u);
}

// ---------------------------------------------------------------------------
// Kernel 1: per-destination max of edge logits (atomic on ordered uint)
// ---------------------------------------------------------------------------
__global__ void __launch_bounds__(256) seg_max_kernel(
    const float* __restrict__ logits, const int* __restrict__ dst,
    unsigned* __restrict__ m_ord, int E) {
    int i = blockIdx.x * 256 + threadIdx.x;
    if (i < E) {
        atomicMax(&m_ord[dst[i]], f32_to_ord(logits[i]));
    }
}

// ---------------------------------------------------------------------------
// Kernel 2: e = exp(logit - m[dst]); z[dst] += e
// sentinel m_ord==0 decodes to NaN -> treated as 0 (matches isfinite guard)
// ---------------------------------------------------------------------------
__global__ void __launch_bounds__(256) exp_z_kernel(
    const float* __restrict__ logits, const int* __restrict__ dst,
    const unsigned* __restrict__ m_ord,
    float* __restrict__ e, float* __restrict__ z, int E) {
    int i = blockIdx.x * 256 + threadIdx.x;
    if (i < E) {
        int d = dst[i];
        unsigned u = m_ord[d];
        unsigned b = (u & 0x80000000u) ? (u & 0x7fffffffu) : ~u;
        bool fin = (b & 0x7f800000u) != 0x7f800000u;
        float m = fin ? __uint_as_float(b) : 0.0f;
        float ev = expf(logits[i] - m);
        e[i] = ev;
        atomicAdd(&z[d], ev);
    }
}

// ---------------------------------------------------------------------------
// Kernel 3: hv = node_feats @ W_proj + b_proj     [N,128] x [128,128]
// 256 threads = 8 wave32; block owns 32 rows; wave w owns cols [16w,16w+16)
// and TWO stacked 16x16 tiles -> each B fragment feeds two WMMAs.
// ---------------------------------------------------------------------------
__global__ void __launch_bounds__(256) proj_kernel(
    const float* __restrict__ A, const float* __restrict__ W,
    const float* __restrict__ bias, float* __restrict__ hv) {
    __shared__ float As[32][132];   // pad 128->132 (stride%64==4): conflict-free columns
    const int rowBase = blockIdx.x * 32;
    const int t = threadIdx.x;
    for (int i = t; i < 32 * 128; i += 256) {
        int r = i >> 7, col = i & 127;
        As[r][col] = A[(rowBase + r) * DFEAT + col];
    }
    __syncthreads();

    const int wave = t >> 5, lane = t & 31;
    const int hf = lane >> 4, l15 = lane & 15;
    const int colBase = wave * 16;

    v8f acc0 = {}, acc1 = {};
    for (int k = 0; k < 128; k += 4) {
        v2f a0, a1, b;
        b.x = W[(k + 2 * hf) * DFEAT + colBase + l15];
        b.y = W[(k + 2 * hf + 1) * DFEAT + colBase + l15];
        a0.x = As[l15][k + 2 * hf];
        a0.y = As[l15][k + 2 * hf + 1];
        a1.x = As[16 + l15][k + 2 * hf];
        a1.y = As[16 + l15][k + 2 * hf + 1];
        acc0 = wmma_f32_k4(a0, b, acc0);
        acc1 = wmma_f32_k4(a1, b, acc1);
    }
    float bc = bias[colBase + l15];
    for (int j = 0; j < 8; ++j) {
        int r = j + 8 * hf;
        hv[(rowBase + r) * DFEAT + colBase + l15]      = acc0[j] + bc;
        hv[(rowBase + 16 + r) * DFEAT + colBase + l15] = acc1[j] + bc;
    }
}

// ---------------------------------------------------------------------------
// Kernel 4: c[dst] += (e/z[dst]) * hv[src]   (one wave per edge, float4 lanes)
// ---------------------------------------------------------------------------
__global__ void __launch_bounds__(256) aggregate_kernel(
    const float* __restrict__ hv, const float* __restrict__ e,
    const float* __restrict__ z, const int* __restrict__ src,
    const int* __restrict__ dst, float* __restrict__ c, int E) {
    int edge = blockIdx.x * 8 + (threadIdx.x >> 5);
    if (edge >= E) return;
    int lane = threadIdx.x & 31;
    int s = src[edge], d = dst[edge];
    float a = e[edge] / z[d];
    const float4* hs = (const float4*)(hv + (size_t)s * DFEAT);
    float4 v = hs[lane];
    float* cd = c + (size_t)d * DFEAT + lane * 4;
    atomicAdd(cd + 0, v.x * a);
    atomicAdd(cd + 1, v.y * a);
    atomicAdd(cd + 2, v.z * a);
    atomicAdd(cd + 3, v.w * a);
}

// ---------------------------------------------------------------------------
// Kernel 5: fused tail MLP over 32-row tiles
//   x = [elu(c), node_feats]  (32x256 in LDS)
//   h = relu(x @ W1 + b1)     (32x128 in LDS)
//   out = relu(h @ W2 + b2)
// ---------------------------------------------------------------------------
__global__ void __launch_bounds__(256) mlp_kernel(
    const float* __restrict__ c, const float* __restrict__ nf,
    const float* __restrict__ W1, const float* __restrict__ b1,
    const float* __restrict__ W2, const float* __restrict__ b2,
    float* __restrict__ out) {
    __shared__ float Xs[32][260];   // 256 padded to 260
    __shared__ float Hs[32][132];   // 128 padded to 132
    const int rowBase = blockIdx.x * 32;
    const int t = threadIdx.x;

    for (int i = t; i < 32 * 256; i += 256) {
        int r = i >> 8, col = i & 255;
        float v;
        if (col < 128) {
            float cv = c[(rowBase + r) * DFEAT + col];
            v = cv > 0.0f ? cv : expm1f(cv);           // elu, alpha=1
        } else {
            v = nf[(rowBase + r) * DFEAT + (col - 128)];
        }
        Xs[r][col] = v;
    }
    __syncthreads();

    const int wave = t >> 5, lane = t & 31;
    const int hf = lane >> 4, l15 = lane & 15;
    const int colBase = wave * 16;

    v8f acc0 = {}, acc1 = {};
    for (int k = 0; k < 256; k += 4) {
        v2f a0, a1, b;
        b.x = W1[(k + 2 * hf) * DFEAT + colBase + l15];
        b.y = W1[(k + 2 * hf + 1) * DFEAT + colBase + l15];
        a0.x = Xs[l15][k + 2 * hf];
        a0.y = Xs[l15][k + 2 * hf + 1];
        a1.x = Xs[16 + l15][k + 2 * hf];
        a1.y = Xs[16 + l15][k + 2 * hf + 1];
        acc0 = wmma_f32_k4(a0, b, acc0);
        acc1 = wmma_f32_k4(a1, b, acc1);
    }
    float bb1 = b1[colBase + l15];
    for (int j = 0; j < 8; ++j) {
        int r = j + 8 * hf;
        float v0 = acc0[j] + bb1;
        float v1 = acc1[j] + bb1;
        Hs[r][colBase + l15]      = v0 > 0.0f ? v0 : 0.0f;   // relu
        Hs[16 + r][colBase + l15] = v1 > 0.0f ? v1 : 0.0f;
    }
    __syncthreads();

    v8f acc2 = {}, acc3 = {};
    for (int k = 0; k < 128; k += 4) {
        v2f a0, a1, b;
        b.x = W2[(k + 2 * hf) * DFEAT + colBase + l15];
        b.y = W2[(k + 2 * hf + 1) * DFEAT + colBase + l15];
        a0.x = Hs[l15][k + 2 * hf];
        a0.y = Hs[l15][k + 2 * hf + 1];
        a1.x = Hs[16 + l15][k + 2 * hf];
        a1.y = Hs[16 + l15][k + 2 * hf + 1];
        acc2 = wmma_f32_k4(a0, b, acc2);
        acc3 = wmma_f32_k4(a1, b, acc3);
    }
    float bb2 = b2[colBase + l15];
    for (int j = 0; j < 8; ++j) {
        int r = j + 8 * hf;
        float v0 = acc2[j] + bb2;
        float v1 = acc3[j] + bb2;
        out[(rowBase + r) * DFEAT + colBase + l15]      = v0 > 0.0f ? v0 : 0.0f;
        out[(rowBase + 16 + r) * DFEAT + colBase + l15] = v1 > 0.0f ? v1 : 0.0f;
    }
}

// ---------------------------------------------------------------------------
extern "C" void kernel_launch(void* const* d_in, const int* in_sizes, int n_in,
                              void* d_out, int out_size, void* d_ws, size_t ws_size,
                              hipStream_t stream) {
    (void)in_sizes; (void)n_in; (void)out_size; (void)ws_size;

    const float* node_feats  = (const float*)d_in[0];
    const float* edge_logits = (const float*)d_in[1];
    const float* W_proj      = (const float*)d_in[2];
    const float* b_proj      = (const float*)d_in[3];
    const float* W1          = (const float*)d_in[4];
    const float* b1          = (const float*)d_in[5];
    const float* W2          = (const float*)d_in[6];
    const float* b2          = (const float*)d_in[7];
    const int*   src         = (const int*)d_in[8];
    const int*   dst         = (const int*)d_in[9];
    float*       out         = (float*)d_out;

    const long N = N_NODES, E = N_EDGES, D = DFEAT;
    char* ws = (char*)d_ws;
    unsigned* m_ord = (unsigned*)ws;                               // N
    float*    z     = (float*)(ws + N * 4);                        // N
    float*    c     = (float*)(ws + 2 * N * 4);                    // N*D
    float*    e     = (float*)(ws + (2 * N + N * D) * 4);          // E
    float*    hv    = (float*)(ws + (2 * N + N * D + E) * 4);      // N*D

    // m_ord sentinel (0 < ord(-inf)), z = 0, c = 0 -> single contiguous memset
    hipMemsetAsync(ws, 0, (size_t)(2 * N + N * D) * 4, stream);

    seg_max_kernel<<<(N_EDGES + 255) / 256, 256, 0, stream>>>(edge_logits, dst, m_ord, N_EDGES);
    exp_z_kernel  <<<(N_EDGES + 255) / 256, 256, 0, stream>>>(edge_logits, dst, m_ord, e, z, N_EDGES);
    proj_kernel   <<<N_NODES / 32, 256, 0, stream>>>(node_feats, W_proj, b_proj, hv);
    aggregate_kernel<<<(N_EDGES + 7) / 8, 256, 0, stream>>>(hv, e, z, src, dst, c, N_EDGES);
    mlp_kernel    <<<N_NODES / 32, 256, 0, stream>>>(c, node_feats, W1, b1, W2, b2, out);
}